// MultiHypothesisTracker_19851338842404
// MI455X (gfx1250) — compile-verified
//
#include <hip/hip_runtime.h>
#include <hip/hip_bf16.h>
#include <math.h>

// ---------------------------------------------------------------------------
// MultiHypothesisTracker on MI455X (gfx1250, wave32).
// Heavy GEMMs run on V_WMMA_F32_16X16X4_F32 (the only fp32 WMMA on CDNA5).
// Block tile 128x128, wave tile 32x64 (8 WMMA accumulators per wave) to keep
// LDS bytes-per-WMMA low; W staged K-pair-interleaved so every B fragment is
// one aligned ds_load_b64 (no VGPR repacking movs).
// ---------------------------------------------------------------------------

typedef __attribute__((ext_vector_type(2))) float v2f;
typedef __attribute__((ext_vector_type(8))) float v8f;

#define B_ROWS 8192
#define HDIM   1024
#define MHYP   4

#define BM 128
#define BN 128
#define BK 32
#define SA_STRIDE 36          // 32 + 4 pad (16B-aligned rows)
#define SW_STRIDE 264         // 2*128 + 8 pad (8B-aligned, conflict-free)

__device__ __forceinline__ float wave_reduce_sum(float v) {
    #pragma unroll
    for (int off = 16; off > 0; off >>= 1)
        v += __shfl_down(v, off, 32);
    return v;
}

__device__ __forceinline__ float gelu_erf(float x) {
    return 0.5f * x * (1.0f + erff(x * 0.70710678118654752f));
}

// ---------------------------------------------------------------------------
// out[B x N] = act( A[B x K] @ W[K x N] + bias + rowBias[rowSel[row]] )
// ---------------------------------------------------------------------------
__global__ __launch_bounds__(256)
void gemm_wmma_f32(const float* __restrict__ A, const float* __restrict__ W,
                   const float* __restrict__ bias,
                   const float* __restrict__ rowBias, const int* __restrict__ rowSel,
                   float* __restrict__ out, int K, int N, int applyGelu) {
    __shared__ float sA[BM * SA_STRIDE];        // 128 x 32 (+pad)
    __shared__ float sWp[(BK / 2) * SW_STRIDE]; // 16 x 256 (+pad), K-pair interleaved

    const int tid  = threadIdx.x;
    const int wave = tid >> 5;
    const int lane = tid & 31;
    const int hi   = lane >> 4;    // half-wave: K pair {0,1} vs {2,3}
    const int l15  = lane & 15;
    const int waveM = wave & 3;    // 4 M-subtiles of 32 rows
    const int waveN = wave >> 2;   // 2 N-subtiles of 64 cols
    const int row0 = blockIdx.y * BM;
    const int n0   = blockIdx.x * BN;

    v8f acc[2][4];
    #pragma unroll
    for (int mi = 0; mi < 2; ++mi)
        #pragma unroll
        for (int nj = 0; nj < 4; ++nj)
            acc[mi][nj] = (v8f){};

    for (int k0 = 0; k0 < K; k0 += BK) {
        __syncthreads();
        // ---- Stage A tile: 128x32 floats as float4 (4 per thread) ----
        #pragma unroll
        for (int i = 0; i < 4; ++i) {
            int idx = tid + i * 256;
            int r = idx >> 3, c4 = (idx & 7) << 2;
            float4 v = *reinterpret_cast<const float4*>(
                A + (size_t)(row0 + r) * K + k0 + c4);
            *reinterpret_cast<float4*>(&sA[r * SA_STRIDE + c4]) = v;
        }
        // ---- Stage W tile: 32x128, K-pair interleaved: sWp[kp][2c+j]=W[2kp+j][c]
        #pragma unroll
        for (int i = 0; i < 2; ++i) {
            int idx = tid + i * 256;                 // 0..511
            int kp = idx >> 5, c4 = (idx & 31) << 2; // kp 0..15, c4 0..124
            const float* w0 = W + (size_t)(k0 + 2 * kp)     * N + n0 + c4;
            const float* w1 = W + (size_t)(k0 + 2 * kp + 1) * N + n0 + c4;
            float4 r0 = *reinterpret_cast<const float4*>(w0);
            float4 r1 = *reinterpret_cast<const float4*>(w1);
            float* dst = &sWp[kp * SW_STRIDE + 2 * c4];
            *reinterpret_cast<float2*>(dst + 0) = make_float2(r0.x, r1.x);
            *reinterpret_cast<float2*>(dst + 2) = make_float2(r0.y, r1.y);
            *reinterpret_cast<float2*>(dst + 4) = make_float2(r0.z, r1.z);
            *reinterpret_cast<float2*>(dst + 6) = make_float2(r0.w, r1.w);
        }
        __syncthreads();

        #pragma unroll
        for (int kk = 0; kk < BK; kk += 4) {
            const int kp = (kk >> 1) + hi;   // K pair index for this half-wave
            // A fragments (16x4 each): contiguous K pair -> single b64 LDS load
            const int ar0 = (waveM * 32 + l15) * SA_STRIDE + kk + (hi << 1);
            v2f a0 = *reinterpret_cast<const v2f*>(&sA[ar0]);
            v2f a1 = *reinterpret_cast<const v2f*>(&sA[ar0 + 16 * SA_STRIDE]);
            // B fragments (4x16 each): interleaved pair -> single b64 LDS load
            const float* bp = &sWp[kp * SW_STRIDE + 2 * (waveN * 64 + l15)];
            v2f b0 = *reinterpret_cast<const v2f*>(bp);
            v2f b1 = *reinterpret_cast<const v2f*>(bp + 32);
            v2f b2 = *reinterpret_cast<const v2f*>(bp + 64);
            v2f b3 = *reinterpret_cast<const v2f*>(bp + 96);
            acc[0][0] = __builtin_amdgcn_wmma_f32_16x16x4_f32(false, a0, false, b0, (short)0, acc[0][0], false, false);
            acc[0][1] = __builtin_amdgcn_wmma_f32_16x16x4_f32(false, a0, false, b1, (short)0, acc[0][1], false, false);
            acc[0][2] = __builtin_amdgcn_wmma_f32_16x16x4_f32(false, a0, false, b2, (short)0, acc[0][2], false, false);
            acc[0][3] = __builtin_amdgcn_wmma_f32_16x16x4_f32(false, a0, false, b3, (short)0, acc[0][3], false, false);
            acc[1][0] = __builtin_amdgcn_wmma_f32_16x16x4_f32(false, a1, false, b0, (short)0, acc[1][0], false, false);
            acc[1][1] = __builtin_amdgcn_wmma_f32_16x16x4_f32(false, a1, false, b1, (short)0, acc[1][1], false, false);
            acc[1][2] = __builtin_amdgcn_wmma_f32_16x16x4_f32(false, a1, false, b2, (short)0, acc[1][2], false, false);
            acc[1][3] = __builtin_amdgcn_wmma_f32_16x16x4_f32(false, a1, false, b3, (short)0, acc[1][3], false, false);
        }
    }

    // Epilogue. C/D layout: lane<16 -> N=lane, M=vgpr i; lane>=16 -> M=i+8.
    #pragma unroll
    for (int nj = 0; nj < 4; ++nj) {
        const int col = n0 + waveN * 64 + nj * 16 + l15;
        const float badd = bias ? bias[col] : 0.f;
        #pragma unroll
        for (int mi = 0; mi < 2; ++mi) {
            #pragma unroll
            for (int i = 0; i < 8; ++i) {
                const int row = row0 + waveM * 32 + mi * 16 + hi * 8 + i;
                float rb = 0.f;
                if (rowBias) rb = rowBias[(size_t)rowSel[row] * N + col];
                float x = acc[mi][nj][i] + badd + rb;
                if (applyGelu) x = gelu_erf(x);
                out[(size_t)row * N + col] = x;
            }
        }
    }
}

// ---------------------------------------------------------------------------
// Init: bank row norms (one wave per row) + argmin over bank_scores
// ---------------------------------------------------------------------------
__global__ void init_kernel(const float* __restrict__ bank,
                            const float* __restrict__ bank_scores,
                            int* __restrict__ minidx, float* __restrict__ minscore,
                            float* __restrict__ bknorm) {
    const int wave = threadIdx.x >> 5, lane = threadIdx.x & 31;
    const float* bm = bank + (size_t)wave * HDIM;
    float acc = 0.f;
    for (int k = lane; k < HDIM; k += 32) { float v = bm[k]; acc += v * v; }
    acc = wave_reduce_sum(acc);
    if (lane == 0) bknorm[wave] = fmaxf(sqrtf(acc), 1e-8f);
    if (threadIdx.x == 0) {
        int mi = 0; float ms = bank_scores[0];
        #pragma unroll
        for (int m = 1; m < MHYP; ++m)
            if (bank_scores[m] < ms) { ms = bank_scores[m]; mi = m; }  // first min
        *minidx = mi; *minscore = ms;
    }
}

// ---------------------------------------------------------------------------
// candproj[m][c] = bank[m] . gW1[H+ :, c] + gb1[c]   (4 x 1024)
// ---------------------------------------------------------------------------
__global__ void candproj_kernel(const float* __restrict__ bank,
                                const float* __restrict__ gW1,
                                const float* __restrict__ gb1,
                                float* __restrict__ candproj) {
    const int m = blockIdx.x >> 2;
    const int c = ((blockIdx.x & 3) << 8) + threadIdx.x;
    const float* bm = bank + (size_t)m * HDIM;
    const float* w  = gW1 + (size_t)HDIM * HDIM + c;   // rows H..2H-1
    float acc = gb1[c];
    for (int k = 0; k < HDIM; ++k) acc += bm[k] * w[(size_t)k * HDIM];
    candproj[m * HDIM + c] = acc;
}

// ---------------------------------------------------------------------------
// Row dot:  out[b] = maybe_sigmoid( x[b,:] . w + bias )   (one wave per row)
// ---------------------------------------------------------------------------
__global__ void rowdot_kernel(const float* __restrict__ x, int ld,
                              const float* __restrict__ w,
                              const float* __restrict__ bias,
                              float* __restrict__ out, int sigm) {
    const int wave = threadIdx.x >> 5, lane = threadIdx.x & 31;
    const int b = blockIdx.x * 8 + wave;
    const float* xr = x + (size_t)b * ld;
    float acc = 0.f;
    for (int j = lane; j < ld; j += 32) acc += xr[j] * w[j];
    acc = wave_reduce_sum(acc);
    if (lane == 0) {
        float y = acc + bias[0];
        if (sigm) y = 1.0f / (1.0f + expf(-y));
        out[b] = y;
    }
}

// ---------------------------------------------------------------------------
// Cosine sims vs 4 bank rows -> max_idx / tgt / (merge|do_update) flags
// ---------------------------------------------------------------------------
__global__ void sims_kernel(const float* __restrict__ nh, const float* __restrict__ bank,
                            const float* __restrict__ score,
                            const int* __restrict__ minidx,
                            const float* __restrict__ minscore,
                            const float* __restrict__ bknorm,
                            int* __restrict__ maxidx, int* __restrict__ tgt,
                            int* __restrict__ flags) {
    __shared__ float sB[MHYP * HDIM];
    for (int i = threadIdx.x; i < MHYP * HDIM; i += 256) sB[i] = bank[i];
    __syncthreads();
    const int wave = threadIdx.x >> 5, lane = threadIdx.x & 31;
    const int b = blockIdx.x * 8 + wave;
    const float* r = nh + (size_t)b * HDIM;
    float nn = 0.f, d0 = 0.f, d1 = 0.f, d2 = 0.f, d3 = 0.f;
    for (int k = lane; k < HDIM; k += 32) {
        const float v = r[k];
        nn += v * v;
        d0 += v * sB[k];
        d1 += v * sB[HDIM + k];
        d2 += v * sB[2 * HDIM + k];
        d3 += v * sB[3 * HDIM + k];
    }
    nn = wave_reduce_sum(nn);
    d0 = wave_reduce_sum(d0); d1 = wave_reduce_sum(d1);
    d2 = wave_reduce_sum(d2); d3 = wave_reduce_sum(d3);
    if (lane == 0) {
        const float nhn = fmaxf(sqrtf(nn), 1e-8f);
        float s[MHYP] = { d0 / (nhn * bknorm[0]), d1 / (nhn * bknorm[1]),
                          d2 / (nhn * bknorm[2]), d3 / (nhn * bknorm[3]) };
        int mi = 0; float mv = s[0];
        #pragma unroll
        for (int m = 1; m < MHYP; ++m)
            if (s[m] > mv) { mv = s[m]; mi = m; }  // first max (strict >)
        const int merge = (mv > 0.8f) ? 1 : 0;
        const int du = (score[b] > *minscore) ? 1 : 0;
        maxidx[b] = mi;
        tgt[b] = merge ? mi : *minidx;
        flags[b] = merge | (du << 1);
    }
}

// ---------------------------------------------------------------------------
// Scatter: hyp_out[b,m,:] and sc_out[b,m]   (one block per batch row)
// ---------------------------------------------------------------------------
__global__ void scatter_kernel(const float* __restrict__ nh, const float* __restrict__ bank,
                               const float* __restrict__ bank_scores,
                               const float* __restrict__ g, const float* __restrict__ score,
                               const int* __restrict__ maxidx, const int* __restrict__ tgt,
                               const int* __restrict__ flags,
                               float* __restrict__ hyp, float* __restrict__ scout) {
    const int b = blockIdx.x;
    const int mi = maxidx[b], t = tgt[b], fl = flags[b];
    const int merge = fl & 1, du = (fl >> 1) & 1;
    const float gv = g[b];
    float* hb = hyp + (size_t)b * (MHYP * HDIM);
    const float* nr = nh + (size_t)b * HDIM;
    for (int k = threadIdx.x; k < HDIM; k += 256) {
        const float nhv = nr[k];
        const float cand = bank[(size_t)mi * HDIM + k];
        const float vec = merge ? (gv * nhv + (1.0f - gv) * cand) : nhv;
        #pragma unroll
        for (int m = 0; m < MHYP; ++m) {
            float val = bank[(size_t)m * HDIM + k];
            if (du && m == t) val = vec;
            hb[(size_t)m * HDIM + k] = val;
        }
    }
    if (threadIdx.x < MHYP) {
        const int m = threadIdx.x;
        float sv = bank_scores[m];
        if (du && m == t) {
            const float ns = score[b];
            sv = merge ? fmaxf(ns, bank_scores[t]) : ns;
        }
        scout[(size_t)b * MHYP + m] = sv;
    }
}

// ---------------------------------------------------------------------------
// Fused LayerNorm over H=1024, in place (one block per row)
// ---------------------------------------------------------------------------
__global__ void layernorm_kernel(float* __restrict__ h, const float* __restrict__ gam,
                                 const float* __restrict__ bet) {
    __shared__ float ws1[8], ws2[8];
    const int b = blockIdx.x;
    float* r = h + (size_t)b * HDIM;
    float v[4], s = 0.f, s2 = 0.f;
    #pragma unroll
    for (int i = 0; i < 4; ++i) {
        v[i] = r[threadIdx.x + i * 256];
        s += v[i]; s2 += v[i] * v[i];
    }
    s = wave_reduce_sum(s); s2 = wave_reduce_sum(s2);
    const int wave = threadIdx.x >> 5, lane = threadIdx.x & 31;
    if (lane == 0) { ws1[wave] = s; ws2[wave] = s2; }
    __syncthreads();
    if (threadIdx.x == 0) {
        float ts = 0.f, ts2 = 0.f;
        #pragma unroll
        for (int w = 0; w < 8; ++w) { ts += ws1[w]; ts2 += ws2[w]; }
        ws1[0] = ts; ws2[0] = ts2;
    }
    __syncthreads();
    const float mean = ws1[0] * (1.0f / HDIM);
    const float var  = ws2[0] * (1.0f / HDIM) - mean * mean;
    const float inv  = rsqrtf(var + 1e-5f);
    #pragma unroll
    for (int i = 0; i < 4; ++i) {
        const int c = threadIdx.x + i * 256;
        r[c] = (v[i] - mean) * inv * gam[c] + bet[c];
    }
}

// ---------------------------------------------------------------------------
extern "C" void kernel_launch(void* const* d_in, const int* in_sizes, int n_in,
                              void* d_out, int out_size, void* d_ws, size_t ws_size,
                              hipStream_t stream) {
    const float* nh    = (const float*)d_in[0];
    // d_in[1] = context (unused by the reference)
    const float* bank  = (const float*)d_in[2];
    const float* bsc   = (const float*)d_in[3];
    const float* sW1   = (const float*)d_in[4];
    const float* sb1   = (const float*)d_in[5];
    const float* sW2   = (const float*)d_in[6];
    const float* sb2   = (const float*)d_in[7];
    const float* gW1   = (const float*)d_in[8];
    const float* gb1   = (const float*)d_in[9];
    const float* gW2   = (const float*)d_in[10];
    const float* gb2   = (const float*)d_in[11];
    const float* cW1   = (const float*)d_in[12];
    const float* cb1   = (const float*)d_in[13];
    const float* cW2   = (const float*)d_in[14];
    const float* cb2   = (const float*)d_in[15];
    const float* ln_g  = (const float*)d_in[16];
    const float* ln_b  = (const float*)d_in[17];

    float* out       = (float*)d_out;
    float* combined  = out;                                   // [B,H]
    float* hyp       = out + (size_t)B_ROWS * HDIM;           // [B,M,H]
    float* scout     = hyp + (size_t)B_ROWS * MHYP * HDIM;    // [B,M]

    float* wsf       = (float*)d_ws;
    float* score     = wsf;                  // B
    float* gbuf      = wsf + 8192;           // B
    int*   maxidx    = (int*)(wsf + 16384);  // B
    int*   tgt       = (int*)(wsf + 24576);  // B
    int*   flags     = (int*)(wsf + 32768);  // B
    int*   minidx    = (int*)(wsf + 40960);
    float* minscore  = wsf + 40961;
    float* bknorm    = wsf + 40962;          // 4
    float* candproj  = wsf + 40976;          // 4*1024
    float* big       = wsf + 65536;          // reused: h1s[B,512] / h1g[B,1024] / h1c[B,2048]

    const dim3 blk(256);

    // 0) bank norms + argmin(bank_scores)
    init_kernel<<<1, 128, 0, stream>>>(bank, bsc, minidx, minscore, bknorm);
    // 0b) gate candidate projections (tiny GEMM: 4x1024x1024)
    candproj_kernel<<<16, 256, 0, stream>>>(bank, gW1, gb1, candproj);

    // 1) scorer GEMM: gelu(nh @ sW1 + sb1) -> big[B,512]
    gemm_wmma_f32<<<dim3(512 / BN, B_ROWS / BM), blk, 0, stream>>>(
        nh, sW1, sb1, nullptr, nullptr, big, HDIM, 512, 1);
    // 2) new_score = big . sW2 + sb2
    rowdot_kernel<<<B_ROWS / 8, 256, 0, stream>>>(big, 512, sW2, sb2, score, 0);

    // 3) cosine similarities -> max_idx / tgt / flags
    sims_kernel<<<B_ROWS / 8, 256, 0, stream>>>(nh, bank, score, minidx, minscore,
                                                bknorm, maxidx, tgt, flags);

    // 4) gate GEMM: gelu(nh @ gW1_top + candproj[max_idx]) -> big[B,1024]
    //    (cand half of the concat folded into the row-indexed bias)
    gemm_wmma_f32<<<dim3(HDIM / BN, B_ROWS / BM), blk, 0, stream>>>(
        nh, gW1, nullptr, candproj, maxidx, big, HDIM, HDIM, 1);
    // 5) g = sigmoid(big . gW2 + gb2)
    rowdot_kernel<<<B_ROWS / 8, 256, 0, stream>>>(big, HDIM, gW2, gb2, gbuf, 1);

    // 6) scatter -> hyp_out[B,M,H] (also the combiner input) + sc_out[B,M]
    scatter_kernel<<<B_ROWS, 256, 0, stream>>>(nh, bank, bsc, gbuf, score,
                                               maxidx, tgt, flags, hyp, scout);

    // 7) combiner GEMM1: gelu(hyp.flat @ cW1 + cb1) -> big[B,2048]
    gemm_wmma_f32<<<dim3(2048 / BN, B_ROWS / BM), blk, 0, stream>>>(
        hyp, cW1, cb1, nullptr, nullptr, big, MHYP * HDIM, 2048, 1);
    // 8) combiner GEMM2: big @ cW2 + cb2 -> combined[B,1024]
    gemm_wmma_f32<<<dim3(HDIM / BN, B_ROWS / BM), blk, 0, stream>>>(
        big, cW2, cb2, nullptr, nullptr, combined, 2048, HDIM, 0);
    // 9) LayerNorm in place
    layernorm_kernel<<<B_ROWS, 256, 0, stream>>>(combined, ln_g, ln_b);
}